// IntensityFunction_87222195847186
// MI455X (gfx1250) — compile-verified
//
#include <hip/hip_runtime.h>
#include <math.h>

// ---------------------------------------------------------------------------
// IntensityFunction for MI455X (gfx1250, wave32).
//   N=32 stocks, S=128 seq, D=256 model dim.
// Phase 1: three fused Linears via V_WMMA_F32_16X16X4_F32 (fp32-exact matrix
//          path; workload is bandwidth bound so low precision buys nothing),
//          + GELU/GELU/softplus10 epilogues -> mu, eta, gamma in workspace.
// Phase 2: causal decay sum via exp->geometric-power recurrence (2 exp + 15
//          mul per 16 outputs instead of 16 exp), fused final softplus.
// ---------------------------------------------------------------------------

typedef __attribute__((ext_vector_type(2))) float v2f;
typedef __attribute__((ext_vector_type(8))) float v8f;

#define NS   32
#define SEQ  128
#define DM   256
#define MROWS (NS * SEQ)          // 4096 rows of the flattened (n,s) GEMM

__device__ __forceinline__ float gelu_exact(float z) {
    // 0.5 * z * (1 + erf(z / sqrt(2)))  -- matches jax.nn.gelu(approximate=False)
    return 0.5f * z * (1.0f + erff(z * 0.70710678118654752440f));
}

__device__ __forceinline__ float softplus1(float v) {
    // softplus with overflow guard; exp underflow -> log1p(0) = 0 is exact
    return (v > 20.0f) ? v : log1pf(expf(v));
}

// ---------------------------------------------------------------------------
// Phase 1: out[m, e] = act( sum_k x[m,k] * W[e,k] + b[e] ) for 3 weight sets.
// One wave computes one 16x16 (M x E) tile; K-loop steps 4 with f32 WMMA.
// ISA f32 A-frag (16x4): lane l holds A[l&15][k0 + 2*(l>>4) + {0,1}] (v2f).
// ISA f32 B-frag (4x16): lane l holds B[2*(l>>4)+{0,1}][l&15]
//                        = W[e0+(l&15)][k0 + 2*(l>>4) + {0,1}]  (v2f).
// C/D 16x16 f32: VGPR v, lane l -> (row = v + 8*(l>>4), col = l&15).
// ---------------------------------------------------------------------------
__global__ __launch_bounds__(256)
void fused_linear3_wmma(const float* __restrict__ x,
                        const float* __restrict__ Wc, const float* __restrict__ bc,
                        const float* __restrict__ Ws, const float* __restrict__ bs,
                        const float* __restrict__ Wd, const float* __restrict__ bd,
                        float* __restrict__ mu,
                        float* __restrict__ eta,
                        float* __restrict__ gam) {
    const int tid  = threadIdx.x;
    const int wave = tid >> 5;
    const int lane = tid & 31;
    const int half = lane >> 4;    // 0: K pair {0,1} rows 0..7 | 1: K pair {2,3} rows 8..15
    const int lr   = lane & 15;

    // 4096 wave-tiles: 256 M-tiles x 16 E-tiles; 8 waves per block -> 512 blocks
    const int tileId = blockIdx.x * 8 + wave;
    const int e0 = (tileId & 15) << 4;
    const int m0 = (tileId >> 4) << 4;

    const float* xrow  = x  + (size_t)(m0 + lr) * DM + 2 * half;
    const float* w1row = Wc + (size_t)(e0 + lr) * DM + 2 * half;
    const float* w2row = Ws + (size_t)(e0 + lr) * DM + 2 * half;
    const float* w3row = Wd + (size_t)(e0 + lr) * DM + 2 * half;

    v8f c1 = {};
    v8f c2 = {};
    v8f c3 = {};

    for (int k0 = 0; k0 < DM; k0 += 4) {
        // hint the streaming x rows ahead (gfx1250 global_prefetch_b8)
        __builtin_prefetch(xrow + k0 + 64, 0, 0);

        v2f a  = *(const v2f*)(xrow  + k0);
        v2f b1 = *(const v2f*)(w1row + k0);
        v2f b2 = *(const v2f*)(w2row + k0);
        v2f b3 = *(const v2f*)(w3row + k0);

        // D = A x B + C; reuse the A fragment across all three weight matrices
        c1 = __builtin_amdgcn_wmma_f32_16x16x4_f32(false, a, false, b1, (short)0, c1, false, false);
        c2 = __builtin_amdgcn_wmma_f32_16x16x4_f32(false, a, false, b2, (short)0, c2, false, false);
        c3 = __builtin_amdgcn_wmma_f32_16x16x4_f32(false, a, false, b3, (short)0, c3, false, false);
    }

    const float biasC = bc[e0 + lr];
    const float biasS = bs[e0 + lr];
    const float biasD = bd[e0 + lr];

#pragma unroll
    for (int v = 0; v < 8; ++v) {
        const int row = m0 + v + 8 * half;
        const int col = e0 + lr;
        const size_t idx = (size_t)row * DM + col;

        mu[idx]  = gelu_exact(c1[v] + biasC);
        eta[idx] = gelu_exact(c2[v] + biasS);

        // gamma = softplus(10*z)/10
        const float z10 = (c3[v] + biasD) * 10.0f;
        const float sp  = (z10 > 20.0f) ? z10 : log1pf(expf(z10));
        gam[idx] = sp * 0.1f;
    }
}

// ---------------------------------------------------------------------------
// Phase 2: influence[n,i,d] = sum_{j<=i} exp(-gamma[n,j,d] * dt(i,j)),
//          dt(i,j) = i-j (j<i), 1e-10 (j==i).
// Block = (n, 16-row i-tile), thread = d (coalesced). For each source column
// j we pay 2 exps (e0 = exp(-g*(i0-j)), m = exp(-g)) then run the geometric
// recurrence e *= m across the 16 rows of the tile.
// Fused epilogue: out = softplus(mu + (eta-mu)*influence).
// ---------------------------------------------------------------------------
__global__ __launch_bounds__(256)
void decay_sum_softplus(const float* __restrict__ mu,
                        const float* __restrict__ eta,
                        const float* __restrict__ gam,
                        float* __restrict__ out) {
    const int n  = blockIdx.x >> 3;          // 32 stocks
    const int i0 = (blockIdx.x & 7) << 4;    // 8 i-tiles of 16 rows
    const int d  = threadIdx.x;              // 256 channels

    const float* gcol = gam + (size_t)n * SEQ * DM + d;

    float acc[16];
#pragma unroll
    for (int k = 0; k < 16; ++k) acc[k] = 0.0f;

    // Columns fully below the tile: contribute to all 16 rows.
    for (int j = 0; j < i0; ++j) {
        const float g = gcol[(size_t)j * DM];
        const float m = expf(-g);
        float e = expf(-g * (float)(i0 - j));
#pragma unroll
        for (int k = 0; k < 16; ++k) { acc[k] += e; e *= m; }
    }

    // Columns inside the tile: diagonal term + partial geometric tail.
#pragma unroll
    for (int jj = 0; jj < 16; ++jj) {
        const float g = gcol[(size_t)(i0 + jj) * DM];
        acc[jj] += expf(-g * 1e-10f);        // delta_t on the diagonal
        const float m = expf(-g);
        float e = m;
        for (int k = jj + 1; k < 16; ++k) { acc[k] += e; e *= m; }
    }

#pragma unroll
    for (int k = 0; k < 16; ++k) {
        const size_t idx = ((size_t)n * SEQ + (i0 + k)) * DM + d;
        const float m_ = mu[idx];
        const float e_ = eta[idx];
        out[idx] = softplus1(m_ + (e_ - m_) * acc[k]);
    }
}

// ---------------------------------------------------------------------------
extern "C" void kernel_launch(void* const* d_in, const int* in_sizes, int n_in,
                              void* d_out, int out_size, void* d_ws, size_t ws_size,
                              hipStream_t stream) {
    (void)in_sizes; (void)n_in; (void)out_size; (void)ws_size;

    const float* x  = (const float*)d_in[0];
    const float* Wc = (const float*)d_in[1];
    const float* bc = (const float*)d_in[2];
    const float* Ws = (const float*)d_in[3];
    const float* bs = (const float*)d_in[4];
    const float* Wd = (const float*)d_in[5];
    const float* bd = (const float*)d_in[6];
    float* out = (float*)d_out;

    const size_t nel = (size_t)MROWS * DM;   // 1,048,576 elems per tensor
    float* mu  = (float*)d_ws;
    float* eta = mu + nel;
    float* gam = eta + nel;

    // 4096 wave-tiles / 8 waves per 256-thread block -> 512 blocks
    fused_linear3_wmma<<<dim3(512), dim3(256), 0, stream>>>(
        x, Wc, bc, Ws, bs, Wd, bd, mu, eta, gam);

    // 32 stocks x 8 i-tiles -> 256 blocks, 256 threads (one per channel d)
    decay_sum_softplus<<<dim3(256), dim3(256), 0, stream>>>(mu, eta, gam, out);
}